// BayesianMetaPosterior_66408784331040
// MI455X (gfx1250) — compile-verified
//
#include <hip/hip_runtime.h>
#include <hip/hip_bf16.h>

typedef __attribute__((ext_vector_type(2))) float v2f;
typedef __attribute__((ext_vector_type(8))) float v8f;

// 32-lane sum using the matrix unit: D = A x B + 0 with B = ones gives row sums
// of A. Each lane contributes its partial once (a[0]=p, a[1]=0). Summing the 8
// C/D VGPRs gives the low-rows total in lanes 0-15 and the high-rows total in
// lanes 16-31 (rows are constant across the 16 columns); one xor-16 shuffle
// combines them. Result valid in every lane. Requires EXEC all-ones: callers
// invoke this from fully-converged, multiple-of-32 blocks.
__device__ __forceinline__ float wave_sum_wmma(float p) {
    v2f a; a[0] = p;    a[1] = 0.0f;
    v2f b; b[0] = 1.0f; b[1] = 1.0f;
    v8f c = {};
    c = __builtin_amdgcn_wmma_f32_16x16x4_f32(
        /*neg_a=*/false, a, /*neg_b=*/false, b,
        /*c_mod=*/(short)0, c, /*reuse_a=*/false, /*reuse_b=*/false);
    float s = ((c[0] + c[1]) + (c[2] + c[3])) + ((c[4] + c[5]) + (c[6] + c[7]));
    s += __shfl_xor(s, 16, 32);
    return s;
}

// Reduce two accumulators across the block (blockDim.x multiple of 32, <=256).
// Totals end up valid in thread 0.
__device__ __forceinline__ void block_sum2(float& a, float& b) {
    a = wave_sum_wmma(a);
    b = wave_sum_wmma(b);
    __shared__ float sm[2][8];
    const int wid = threadIdx.x >> 5;
    const int lid = threadIdx.x & 31;
    if (lid == 0) { sm[0][wid] = a; sm[1][wid] = b; }
    __syncthreads();
    if (threadIdx.x == 0) {
        const int nw = blockDim.x >> 5;
        float ta = 0.0f, tb = 0.0f;
        for (int w = 0; w < nw; ++w) { ta += sm[0][w]; tb += sm[1][w]; }
        a = ta; b = tb;
    }
}

// Pass 1: stream metamean (sum of squares) and fishers (sum of logs) with
// 128-bit loads; one (ssq, slog) pair per block into d_ws.
__global__ void bayes_partials_kernel(const float* __restrict__ mm,
                                      const float* __restrict__ fish,
                                      int nD, int nF,
                                      float* __restrict__ partials) {
    const int gid    = blockIdx.x * blockDim.x + threadIdx.x;
    const int stride = gridDim.x * blockDim.x;

    float ssq = 0.0f;
    {
        const int nD4 = nD >> 2;
        const float4* __restrict__ mm4 = (const float4*)mm;
        for (int i = gid; i < nD4; i += stride) {
            __builtin_prefetch(&mm4[i + 4 * stride], 0, 0);  // global_prefetch_b8
            float4 v = mm4[i];
            ssq += v.x * v.x + v.y * v.y + v.z * v.z + v.w * v.w;
        }
        for (int i = (nD4 << 2) + gid; i < nD; i += stride) {
            float v = mm[i];
            ssq += v * v;
        }
    }

    float slog = 0.0f;
    {
        const int nF4 = nF >> 2;
        const float4* __restrict__ f4 = (const float4*)fish;
        for (int i = gid; i < nF4; i += stride) {
            __builtin_prefetch(&f4[i + 4 * stride], 0, 0);
            float4 v = f4[i];
            slog += (__logf(v.x) + __logf(v.y)) + (__logf(v.z) + __logf(v.w));
        }
        for (int i = (nF4 << 2) + gid; i < nF; i += stride) {
            slog += __logf(fish[i]);
        }
    }

    block_sum2(ssq, slog);
    if (threadIdx.x == 0) {
        partials[2 * blockIdx.x + 0] = ssq;
        partials[2 * blockIdx.x + 1] = slog;
    }
}

// Pass 2: single block folds the per-block partials and applies the closed form.
__global__ void bayes_finalize_kernel(const float* __restrict__ partials,
                                      int nblocks, int nD, int nF,
                                      float* __restrict__ out) {
    float ssq = 0.0f, slog = 0.0f;
    for (int i = threadIdx.x; i < nblocks; i += blockDim.x) {
        ssq  += partials[2 * i + 0];
        slog += partials[2 * i + 1];
    }
    block_sum2(ssq, slog);
    if (threadIdx.x == 0) {
        const double log2pi      = 1.8378770664093454;   // log(2*pi)
        const double half_log2pi = 0.9189385332046727;
        const double log_sigma   = -2.302585092994046;   // log(0.1)
        const double Mm1         = 2.0;                  // (M - 1)
        // (M-1) * [ 0.5/sigma^2 * sum(x^2) + D*(log sigma + 0.5*log 2pi) ]
        double loss = Mm1 * (50.0 * (double)ssq
                             + (double)nD * (log_sigma + half_log2pi));
        // -0.5*M*D*log 2pi + 0.5*sum(log f), with M*D == nF
        loss += -half_log2pi * (double)nF + 0.5 * (double)slog;
        out[0] = (float)loss;
    }
}

extern "C" void kernel_launch(void* const* d_in, const int* in_sizes, int n_in,
                              void* d_out, int out_size, void* d_ws, size_t ws_size,
                              hipStream_t stream) {
    const float* metamean = (const float*)d_in[0];
    // d_in[1] (means) is mathematically dead: (means - means) == 0.
    const float* fishers  = (const float*)d_in[2];
    const int nD = in_sizes[0];
    const int nF = in_sizes[2];

    int nblocks = 1024;
    const size_t need_per_block = 2 * sizeof(float);
    if ((size_t)nblocks * need_per_block > ws_size) {
        nblocks = (int)(ws_size / need_per_block);
        if (nblocks < 1) nblocks = 1;
    }
    float* partials = (float*)d_ws;

    bayes_partials_kernel<<<nblocks, 256, 0, stream>>>(metamean, fishers, nD, nF, partials);
    bayes_finalize_kernel<<<1, 256, 0, stream>>>(partials, nblocks, nD, nF, (float*)d_out);
}